// RWKV_RNN_8272107012230
// MI455X (gfx1250) — compile-verified
//
#include <hip/hip_runtime.h>
#include <math.h>

#define NN 1024
#define FF 4096
#define LL 24
#define VV 50277

typedef __attribute__((ext_vector_type(2))) float v2f;
typedef __attribute__((ext_vector_type(8))) float v8f;

// ---------------------------------------------------------------------------
// x0 = (emb[ctx[-1]]*4 + emb[ctx[-2]]) * 0.2
// ---------------------------------------------------------------------------
__global__ void embed_kernel(const int* __restrict__ ctx,
                             const float* __restrict__ emb,
                             float* __restrict__ x) {
  int i = blockIdx.x * blockDim.x + threadIdx.x;
  if (i < NN) {
    int c1 = ctx[1], c0 = ctx[0];
    x[i] = (emb[(size_t)c1 * NN + i] * 4.0f + emb[(size_t)c0 * NN + i]) * 0.2f;
  }
}

// ---------------------------------------------------------------------------
// LayerNorm over N=1024 + up to three token-mixes against prev-state row.
// One block of 1024 threads.
// ---------------------------------------------------------------------------
__global__ void ln_mix_kernel(const float* __restrict__ xin,
                              const float* __restrict__ w,
                              const float* __restrict__ b,
                              float* __restrict__ out_x,
                              const float* __restrict__ mix0,
                              const float* __restrict__ mix1,
                              const float* __restrict__ mix2,
                              const float* __restrict__ sprev,
                              float* __restrict__ out0,
                              float* __restrict__ out1,
                              float* __restrict__ out2) {
  __shared__ float red[NN];
  int i = threadIdx.x;
  float v = xin[i];
  red[i] = v;
  __syncthreads();
  for (int s = NN / 2; s > 0; s >>= 1) {
    if (i < s) red[i] += red[i + s];
    __syncthreads();
  }
  float mu = red[0] * (1.0f / NN);
  __syncthreads();
  float d = v - mu;
  red[i] = d * d;
  __syncthreads();
  for (int s = NN / 2; s > 0; s >>= 1) {
    if (i < s) red[i] += red[i + s];
    __syncthreads();
  }
  float var = red[0] * (1.0f / NN);
  float xl = d * rsqrtf(var + 1e-5f) * w[i] + b[i];
  out_x[i] = xl;
  float sp = sprev ? sprev[i] : 0.0f;
  if (mix0) { float m = mix0[i]; out0[i] = xl * m + sp * (1.0f - m); }
  if (mix1) { float m = mix1[i]; out1[i] = xl * m + sp * (1.0f - m); }
  if (mix2) { float m = mix2[i]; out2[i] = xl * m + sp * (1.0f - m); }
}

// ---------------------------------------------------------------------------
// WKV linear-attention recurrence (elementwise over N).
// Writes new state rows 2,3,4 and rab = r*a/b.
// ---------------------------------------------------------------------------
__global__ void wkv_kernel(const float* __restrict__ kb_,
                           const float* __restrict__ vb,
                           const float* __restrict__ rb,
                           const float* __restrict__ tf,
                           const float* __restrict__ td,
                           const float* __restrict__ s_in,
                           float* __restrict__ s_out,
                           float* __restrict__ rab) {
  int i = blockIdx.x * blockDim.x + threadIdx.x;
  if (i >= NN) return;
  float aa = s_in[2 * NN + i], bb = s_in[3 * NN + i], pp = s_in[4 * NN + i];
  float k = kb_[i], v = vb[i], r = rb[i];
  float ww = tf[i] + k;
  float q = fmaxf(pp, ww);
  float e1 = expf(pp - q), e2 = expf(ww - q);
  float a = e1 * aa + e2 * v;
  float bden = e1 * bb + e2;
  float ww2 = pp + td[i];
  float q2 = fmaxf(ww2, k);
  float f1 = expf(ww2 - q2), f2 = expf(k - q2);
  s_out[2 * NN + i] = f1 * aa + f2 * v;
  s_out[3 * NN + i] = f1 * bb + f2;
  s_out[4 * NN + i] = q2;
  rab[i] = r * a / bden;
}

// ---------------------------------------------------------------------------
// Bandwidth-bound matvec y = f(W x) using V_WMMA_F32_16X16X4_F32.
// Broadcast-B trick: B[k][n] = x[kb+k] for all n, so every D column holds the
// same 16 row-dots. Block = 8 waves, 16 rows; waves split K 8-way, LDS-reduce.
// A-fragment layout (ISA 7.12.2): lanes 0-15 hold rows M=0..15 at K=kb,kb+1;
// lanes 16-31 hold the same rows at K=kb+2,kb+3 -> kofs = (lane>>4)*2.
// Inner body: 16 WMMAs fully unrolled, two accumulators to break the D->C
// RAW chain, one unconditional prefetch per 64-K chunk (no EXEC churn).
// Requires K % 512 == 0 (holds: K is 1024 or 4096).
// MODE: 0 = identity, 1 = sigmoid, 2 = relu^2. Then *mulv, +addv.
// ---------------------------------------------------------------------------
template <int MODE>
__global__ void wmma_matvec_kernel(const float* __restrict__ W,
                                   const float* __restrict__ x,
                                   float* __restrict__ y,
                                   int M, int K,
                                   const float* __restrict__ addv,
                                   const float* __restrict__ mulv) {
  const int lane = threadIdx.x & 31;
  const int wave = threadIdx.x >> 5;   // 0..7
  const int row0 = blockIdx.x * 16;
  int row = row0 + (lane & 15);
  if (row >= M) row = M - 1;  // clamp: duplicate loads, guarded store
  const int kofs = (lane >> 4) << 1;   // 0 or 2
  const int kchunk = K >> 3;           // K split across 8 waves
  const int kbeg = wave * kchunk;
  const int kend = kbeg + kchunk;
  const float* __restrict__ wrow = W + (size_t)row * K;

  v8f acc0 = {};
  v8f acc1 = {};
  for (int kb = kbeg; kb < kend; kb += 64) {
    __builtin_prefetch(wrow + kb + 256, 0, 1);   // unconditional: no branch
#pragma unroll
    for (int t = 0; t < 64; t += 8) {
      v2f a0 = *(const v2f*)(wrow + kb + t + kofs);
      v2f b0 = *(const v2f*)(x + kb + t + kofs);
      acc0 = __builtin_amdgcn_wmma_f32_16x16x4_f32(
          false, a0, false, b0, (short)0, acc0, false, false);
      v2f a1 = *(const v2f*)(wrow + kb + t + 4 + kofs);
      v2f b1 = *(const v2f*)(x + kb + t + 4 + kofs);
      acc1 = __builtin_amdgcn_wmma_f32_16x16x4_f32(
          false, a1, false, b1, (short)0, acc1, false, false);
    }
  }

  __shared__ float red[8][16];
  if (lane == 0) {
    for (int j = 0; j < 8; ++j) red[wave][j] = acc0[j] + acc1[j];       // rows +0..7
  }
  if (lane == 16) {
    for (int j = 0; j < 8; ++j) red[wave][8 + j] = acc0[j] + acc1[j];   // rows +8..15
  }
  __syncthreads();
  if (threadIdx.x < 16) {
    int r = row0 + threadIdx.x;
    if (r < M) {
      float v = 0.0f;
      for (int wv = 0; wv < 8; ++wv) v += red[wv][threadIdx.x];
      if (MODE == 1) v = 1.0f / (1.0f + expf(-v));
      if (MODE == 2) v = (v > 0.0f) ? v * v : 0.0f;
      if (mulv) v *= mulv[r];
      if (addv) v += addv[r];
      y[r] = v;
    }
  }
}

// ---------------------------------------------------------------------------
extern "C" void kernel_launch(void* const* d_in, const int* in_sizes, int n_in,
                              void* d_out, int out_size, void* d_ws, size_t ws_size,
                              hipStream_t stream) {
  const int*   ctx      = (const int*)d_in[0];
  const float* state_in = (const float*)d_in[1];
  const float* emb      = (const float*)d_in[2];
  const float* ln0_w    = (const float*)d_in[3];
  const float* ln0_b    = (const float*)d_in[4];
  const float* ln1_w    = (const float*)d_in[5];
  const float* ln1_b    = (const float*)d_in[6];
  const float* amk      = (const float*)d_in[7];
  const float* amv      = (const float*)d_in[8];
  const float* amr      = (const float*)d_in[9];
  const float* tf       = (const float*)d_in[10];
  const float* td       = (const float*)d_in[11];
  const float* kw       = (const float*)d_in[12];
  const float* vw       = (const float*)d_in[13];
  const float* rw       = (const float*)d_in[14];
  const float* ow       = (const float*)d_in[15];
  const float* ln2_w    = (const float*)d_in[16];
  const float* ln2_b    = (const float*)d_in[17];
  const float* fmk      = (const float*)d_in[18];
  const float* fmr      = (const float*)d_in[19];
  const float* fkw      = (const float*)d_in[20];
  const float* fvw      = (const float*)d_in[21];
  const float* frw      = (const float*)d_in[22];
  const float* lnout_w  = (const float*)d_in[23];
  const float* lnout_b  = (const float*)d_in[24];
  const float* head     = (const float*)d_in[25];

  float* logits    = (float*)d_out;          // V floats
  float* state_out = (float*)d_out + VV;     // 5*L*N floats

  float* ws  = (float*)d_ws;
  float* x   = ws;                // current residual stream
  float* xk  = ws + 1 * NN;
  float* xv  = ws + 2 * NN;
  float* xr  = ws + 3 * NN;
  float* kb  = ws + 4 * NN;
  float* vb  = ws + 5 * NN;
  float* rb  = ws + 6 * NN;       // sigmoid already applied
  float* rab = ws + 7 * NN;       // r*a/b
  float* sx  = ws + 8 * NN;
  float* fxk = ws + 9 * NN;
  float* fxr = ws + 10 * NN;
  float* r2  = ws + 11 * NN;      // sigmoid already applied
  float* kk  = ws + 12 * NN;      // FF floats (relu^2 applied)
  float* xn  = ws + 12 * NN + FF;

  dim3 mvBlock(256);
  auto mvGrid = [](int M) { return dim3((unsigned)((M + 15) / 16)); };

  embed_kernel<<<dim3((NN + 255) / 256), dim3(256), 0, stream>>>(ctx, emb, x);
  ln_mix_kernel<<<1, NN, 0, stream>>>(x, ln0_w, ln0_b, x, nullptr, nullptr,
                                      nullptr, nullptr, nullptr, nullptr, nullptr);

  for (int l = 0; l < LL; ++l) {
    const float* s_in = state_in + (size_t)5 * l * NN;
    float* s_out = state_out + (size_t)5 * l * NN;

    // --- attention half: ln1 + mixes; xl goes straight into new state row 1
    ln_mix_kernel<<<1, NN, 0, stream>>>(x, ln1_w + l * NN, ln1_b + l * NN,
                                        s_out + 1 * NN,
                                        amk + l * NN, amv + l * NN, amr + l * NN,
                                        s_in + 1 * NN, xk, xv, xr);
    wmma_matvec_kernel<0><<<mvGrid(NN), mvBlock, 0, stream>>>(
        kw + (size_t)l * NN * NN, xk, kb, NN, NN, nullptr, nullptr);
    wmma_matvec_kernel<0><<<mvGrid(NN), mvBlock, 0, stream>>>(
        vw + (size_t)l * NN * NN, xv, vb, NN, NN, nullptr, nullptr);
    wmma_matvec_kernel<1><<<mvGrid(NN), mvBlock, 0, stream>>>(
        rw + (size_t)l * NN * NN, xr, rb, NN, NN, nullptr, nullptr);
    wkv_kernel<<<dim3(NN / 256), dim3(256), 0, stream>>>(
        kb, vb, rb, tf + l * NN, td + l * NN, s_in, s_out, rab);
    // sx = x + ow @ rab
    wmma_matvec_kernel<0><<<mvGrid(NN), mvBlock, 0, stream>>>(
        ow + (size_t)l * NN * NN, rab, sx, NN, NN, x, nullptr);

    // --- FFN half: ln2 + mixes; x2 goes straight into new state row 0
    ln_mix_kernel<<<1, NN, 0, stream>>>(sx, ln2_w + l * NN, ln2_b + l * NN,
                                        s_out + 0 * NN,
                                        fmk + l * NN, fmr + l * NN, nullptr,
                                        s_in + 0 * NN, fxk, fxr, nullptr);
    wmma_matvec_kernel<1><<<mvGrid(NN), mvBlock, 0, stream>>>(
        frw + (size_t)l * NN * NN, fxr, r2, NN, NN, nullptr, nullptr);
    wmma_matvec_kernel<2><<<mvGrid(FF), mvBlock, 0, stream>>>(
        fkw + (size_t)l * FF * NN, fxk, kk, FF, NN, nullptr, nullptr);
    // x = sx + r2 * (fvw @ kk)
    wmma_matvec_kernel<0><<<mvGrid(NN), mvBlock, 0, stream>>>(
        fvw + (size_t)l * NN * FF, kk, x, NN, FF, sx, r2);
  }

  ln_mix_kernel<<<1, NN, 0, stream>>>(x, lnout_w, lnout_b, xn, nullptr, nullptr,
                                      nullptr, nullptr, nullptr, nullptr, nullptr);
  wmma_matvec_kernel<0><<<mvGrid(VV), mvBlock, 0, stream>>>(
      head, xn, logits, VV, NN, nullptr, nullptr);
}